// SingleModalFactorization_34419867910916
// MI455X (gfx1250) — compile-verified
//
#include <hip/hip_runtime.h>

// Problem constants (match reference).
#define N_SAMPLES   2048
#define N_CHANNELS  4096
#define N_COMP      4
#define DMIN        (-4)          // shift in [-4,4) -> taps d in [-4,4]
#define NTAPS       9
#define KP          (NTAPS * N_COMP)   // 36 = inner GEMM dimension
#define TILE_M      16
#define TILE_N      128
#define LDSB_STRIDE 144           // 128 data dwords + 16 dword TDM pad per row
#define LDSW_STRIDE 37            // odd stride -> conflict-free column reads

typedef __attribute__((ext_vector_type(2))) float        v2f;
typedef __attribute__((ext_vector_type(8))) float        v8f;
typedef __attribute__((ext_vector_type(4))) unsigned int u32x4;
typedef __attribute__((ext_vector_type(8))) int          i32x8;
typedef __attribute__((ext_vector_type(4))) int          i32x4;

// ---------------------------------------------------------------------------
// Pass 1a: B[j, c] = comp[k, c+d] (zero outside [0,C)), j = (d+4)*4 + k.
// 36*4096 floats = 576 KB -> stays resident in L2 for the whole GEMM.
// ---------------------------------------------------------------------------
__global__ void build_B_kernel(const float* __restrict__ comp,
                               float* __restrict__ B) {
    int idx = blockIdx.x * blockDim.x + threadIdx.x;
    if (idx >= KP * N_CHANNELS) return;
    int j = idx / N_CHANNELS;
    int c = idx - j * N_CHANNELS;
    int k = j & (N_COMP - 1);
    int d = (j >> 2) + DMIN;
    int src = c + d;
    float v = 0.0f;
    if (src >= 0 && src < N_CHANNELS) v = comp[k * N_CHANNELS + src];
    B[idx] = v;
}

// ---------------------------------------------------------------------------
// Pass 1b: W[s, j] = contrib[s,k] * tap-weight(d; shift[s,k]).
// Exactly two nonzeros per (s,k): d==m gets (1-f), d==m+1 gets f.
// ---------------------------------------------------------------------------
__global__ void build_W_kernel(const float* __restrict__ contrib,
                               const float* __restrict__ shift,
                               float* __restrict__ W) {
    int idx = blockIdx.x * blockDim.x + threadIdx.x;
    if (idx >= N_SAMPLES * KP) return;
    int s = idx / KP;
    int j = idx - s * KP;
    int k = j & (N_COMP - 1);
    int d = (j >> 2) + DMIN;
    float sh = shift[s * N_COMP + k];
    float ct = contrib[s * N_COMP + k];
    float mf = floorf(sh);
    float f  = sh - mf;
    int   m  = (int)mf;
    float w = 0.0f;
    if (d == m)          w = ct * (1.0f - f);
    else if (d == m + 1) w = ct * f;
    W[idx] = w;
}

// ---------------------------------------------------------------------------
// Pass 2: out(2048x4096) = W(2048x36) x B(36x4096) via V_WMMA_F32_16X16X4_F32.
// Block = 256 threads = 8 waves; each wave owns one 16x16 output tile,
// block covers 16 rows x 128 cols. B slice staged into LDS by the TDM
// (tensor_load_to_lds) with row padding 128->144 dwords so the two half-wave
// read groups land in disjoint LDS bank sets.
// ---------------------------------------------------------------------------
__global__ void __launch_bounds__(256)
shiftnmf_wmma_kernel(const float* __restrict__ W,
                     const float* __restrict__ B,
                     float* __restrict__ out) {
    __shared__ float ldsB[KP * LDSB_STRIDE];     // 36 * 144 * 4B = 20.25 KB
    __shared__ float ldsW[TILE_M * LDSW_STRIDE]; // 16 * 37  * 4B ~= 2.3 KB

    const int m_base  = blockIdx.y * TILE_M;
    const int n_block = blockIdx.x * TILE_N;

    // ---- TDM: async-stage the 36 x 128 B tile into LDS (one descriptor) ----
    if (threadIdx.x == 0) {
        unsigned long long gaddr =
            (unsigned long long)(const void*)(B + n_block);
        unsigned int laddr =
            (unsigned int)(unsigned long long)(const void*)&ldsB[0];

        u32x4 g0;
        g0[0] = 1u;                              // count=1 (valid), user mode
        g0[1] = laddr;                           // lds_addr (bytes)
        g0[2] = (unsigned int)gaddr;             // global_addr[31:0]
        g0[3] = ((unsigned int)(gaddr >> 32) & 0x01FFFFFFu)
              | (2u << 30);                      // global_addr[56:32] | type=2

        i32x8 g1;
        g1[0] = (2  << 16)                       // data_size = 4 bytes
              | (1  << 20)                       // pad_enable
              | (6  << 22)                       // pad_interval: 128 dwords
              | (15 << 25);                      // pad_amount:   16 dwords
        g1[1] = (N_CHANNELS & 0xFFFF) << 16;     // tensor_dim0 lo16 (=4096)
        g1[2] = (KP & 0xFFFF) << 16;             // dim0 hi16=0 | tensor_dim1=36
        g1[3] = (TILE_N & 0xFFFF) << 16;         // dim1 hi16=0 | tile_dim0=128
        g1[4] = (KP & 0xFFFF);                   // tile_dim1=36 | tile_dim2=0
        g1[5] = N_CHANNELS;                      // tensor_dim0_stride = 4096
        g1[6] = 0;                               // stride hi | dim1_stride lo
        g1[7] = 0;
        i32x4 gz4 = {0, 0, 0, 0};
        i32x8 gz8 = {0, 0, 0, 0, 0, 0, 0, 0};
        // amdgpu-toolchain (clang-23) 6-arg form: g0, g1, g2, g3, extra, cpol
        __builtin_amdgcn_tensor_load_to_lds(g0, g1, gz4, gz4, gz8, 0);
    }

    // Stage the 16 x 36 W tile (tiny) while the TDM streams B.
    for (int i = threadIdx.x; i < TILE_M * KP; i += 256) {
        int r  = i / KP;
        int cw = i - r * KP;
        ldsW[r * LDSW_STRIDE + cw] = W[(m_base + r) * KP + cw];
    }

    __builtin_amdgcn_s_wait_tensorcnt(0);  // issuing wave waits for its TDM op
    __syncthreads();                       // LDS visible to all 8 waves

    const int wid  = threadIdx.x >> 5;     // wave id 0..7 -> N sub-tile
    const int lane = threadIdx.x & 31;
    const int half = lane >> 4;            // 0: lanes 0-15, 1: lanes 16-31
    const int lrow = lane & 15;
    const int nb   = wid * 16;

    // A 16x4 f32: lanes 0-15 carry K={0,1}, lanes 16-31 carry K={2,3}.
    // B 4x16 f32: VGPR0 rows K=0/2 (by half), VGPR1 rows K=1/3.
    v8f acc = {};
#pragma unroll
    for (int t = 0; t < NTAPS; ++t) {
        const int k0 = t * 4 + 2 * half;
        v2f a, b;
        a.x = ldsW[lrow * LDSW_STRIDE + k0];
        a.y = ldsW[lrow * LDSW_STRIDE + k0 + 1];
        b.x = ldsB[(k0)     * LDSB_STRIDE + nb + lrow];
        b.y = ldsB[(k0 + 1) * LDSB_STRIDE + nb + lrow];
        acc = __builtin_amdgcn_wmma_f32_16x16x4_f32(
            /*neg_a=*/false, a, /*neg_b=*/false, b,
            /*c_mod=*/(short)0, acc, /*reuse_a=*/false, /*reuse_b=*/false);
    }

    // D layout: VGPR i -> row m_base + i + 8*half, column from lane group.
    const int col = n_block + nb + lrow;
    float* orow = out + (long)(m_base + 8 * half) * N_CHANNELS + col;
#pragma unroll
    for (int i = 0; i < 8; ++i) {
        orow[(long)i * N_CHANNELS] = acc[i];
    }
}

// ---------------------------------------------------------------------------
extern "C" void kernel_launch(void* const* d_in, const int* in_sizes, int n_in,
                              void* d_out, int out_size, void* d_ws,
                              size_t ws_size, hipStream_t stream) {
    // setup_inputs order: inputs (unused by reference math), components,
    // contributions, shift.
    const float* components    = (const float*)d_in[1];
    const float* contributions = (const float*)d_in[2];
    const float* shift         = (const float*)d_in[3];
    float* out = (float*)d_out;

    float* Bm = (float*)d_ws;                         // 36*4096 f32 = 576 KB
    float* Wm = Bm + (size_t)KP * N_CHANNELS;         // 2048*36 f32 = 288 KB

    build_B_kernel<<<(KP * N_CHANNELS + 255) / 256, 256, 0, stream>>>(
        components, Bm);
    build_W_kernel<<<(N_SAMPLES * KP + 255) / 256, 256, 0, stream>>>(
        contributions, shift, Wm);

    dim3 grid(N_CHANNELS / TILE_N, N_SAMPLES / TILE_M);  // 32 x 128 blocks
    shiftnmf_wmma_kernel<<<grid, 256, 0, stream>>>(Wm, Bm, out);
}